// GraphConvolution_5334349382168
// MI455X (gfx1250) — compile-verified
//
#include <hip/hip_runtime.h>
#include <math.h>

#define N_SRC   50000
#define N_DST   25000
#define N_EDGES 500000
#define NH      512

typedef float v2f __attribute__((ext_vector_type(2)));
typedef float v8f __attribute__((ext_vector_type(8)));

// ---------------- CSR build ----------------

__global__ void gc_zero_kernel(int* p, int n) {
    int i = blockIdx.x * blockDim.x + threadIdx.x;
    if (i < n) p[i] = 0;
}

__global__ void gc_hist_kernel(const int* __restrict__ rows, int* __restrict__ row_ptr) {
    int e = blockIdx.x * blockDim.x + threadIdx.x;
    if (e < N_EDGES) atomicAdd(&row_ptr[1 + rows[e]], 1);
}

// single-block inclusive scan over row_ptr[0..N_DST] (25001 elements)
__global__ void gc_scan_kernel(int* __restrict__ row_ptr) {
    __shared__ int buf[1024];
    __shared__ int carry_s;
    const int tid = threadIdx.x;
    if (tid == 0) carry_s = 0;
    __syncthreads();
    for (int base = 0; base < N_DST + 1; base += 1024) {
        int i = base + tid;
        int v = (i <= N_DST) ? row_ptr[i] : 0;
        buf[tid] = v;
        __syncthreads();
        for (int off = 1; off < 1024; off <<= 1) {
            int t = (tid >= off) ? buf[tid - off] : 0;
            __syncthreads();
            buf[tid] += t;
            __syncthreads();
        }
        int out = buf[tid] + carry_s;
        if (i <= N_DST) row_ptr[i] = out;
        __syncthreads();
        if (tid == 1023) carry_s = out;
        __syncthreads();
    }
}

__global__ void gc_scatter_kernel(const int* __restrict__ rows,
                                  const int* __restrict__ row_ptr,
                                  int* __restrict__ fill,
                                  int* __restrict__ eidx) {
    int e = blockIdx.x * blockDim.x + threadIdx.x;
    if (e < N_EDGES) {
        int r = rows[e];
        int pos = atomicAdd(&fill[r], 1);
        eidx[row_ptr[r] + pos] = e;
    }
}

// ---------------- gather SpMM: agg[r] = sum_e val[e]*x[col[e]] ----------------
// one block (4 waves, 128 threads) per destination row; no atomics

__global__ void gc_spmm_kernel(const float* __restrict__ x,
                               const int* __restrict__ cols,
                               const float* __restrict__ vals,
                               const int* __restrict__ row_ptr,
                               const int* __restrict__ eidx,
                               float* __restrict__ agg) {
    __shared__ float part[4 * NH];
    const int r = blockIdx.x;
    const int wave = threadIdx.x >> 5;
    const int lane = threadIdx.x & 31;
    const int beg = row_ptr[r];
    const int end = row_ptr[r + 1];

    float4 acc[4];
    #pragma unroll
    for (int c = 0; c < 4; ++c) { acc[c].x = 0.f; acc[c].y = 0.f; acc[c].z = 0.f; acc[c].w = 0.f; }

    for (int e = beg + wave; e < end; e += 4) {
        int eid = eidx[e];
        float v = vals[eid];
        const float4* xr = (const float4*)(x + (size_t)cols[eid] * NH);
        #pragma unroll
        for (int c = 0; c < 4; ++c) {
            float4 xv = xr[c * 32 + lane];
            acc[c].x += v * xv.x; acc[c].y += v * xv.y;
            acc[c].z += v * xv.z; acc[c].w += v * xv.w;
        }
    }
    float4* pw = (float4*)&part[wave * NH];
    #pragma unroll
    for (int c = 0; c < 4; ++c) pw[c * 32 + lane] = acc[c];
    __syncthreads();

    const int t = threadIdx.x;               // 0..127 float4 slots
    float4 s;
    s.x = 0.f; s.y = 0.f; s.z = 0.f; s.w = 0.f;
    #pragma unroll
    for (int w = 0; w < 4; ++w) {
        float4 p = ((const float4*)&part[w * NH])[t];
        s.x += p.x; s.y += p.y; s.z += p.z; s.w += p.w;
    }
    ((float4*)(agg + (size_t)r * NH))[t] = s;
}

// ---------------- fused WMMA-f32 GEMM + bias + momentum + ELU + layernorm ----------------
// block = 256 threads (8 waves); computes rows [bm*32, bm*32+32) x all 512 cols.
// wave w owns 4 column tiles (cols [w*64, w*64+64)) x 2 row tiles; each B frag
// feeds two WMMAs -> 1.5 DS dwords per WMMA.
// Staging is split into load-all / store-all phases so 8 global_load_b128 stay
// in flight instead of serializing on loadcnt 0 per chunk.
// LDS: Bs = W chunk 16x(512+16); As = A chunk 32x(16+1); Os (16x516 slab, two
// epilogue passes) aliases Bs.

#define KC        16
#define BS_STRIDE 528            // 512 + 16 pad -> half-wave frag reads hit disjoint banks
#define AS_STRIDE 17
#define OS_STRIDE 516

__global__ void __launch_bounds__(256)
gc_gemm_fused_kernel(const float* __restrict__ agg,
                     const float* __restrict__ W,
                     const float* __restrict__ bias,
                     const int* __restrict__ sampled_nodes,
                     const float* __restrict__ y_buf,
                     const float* __restrict__ scale,
                     const float* __restrict__ offset,
                     float* __restrict__ out) {
    __shared__ __align__(16) char smem[16 * BS_STRIDE * 4 + 32 * AS_STRIDE * 4 + 128];
    float* Bs = (float*)smem;                                   // 33792 B
    float* As = (float*)(smem + 16 * BS_STRIDE * 4);            // 2176 B
    int*   sn_s = (int*)(smem + 16 * BS_STRIDE * 4 + 32 * AS_STRIDE * 4);  // 32 ints
    float* Os = (float*)smem;                                   // aliases Bs (33024 B used)

    const int bm   = blockIdx.x;
    const int tid  = threadIdx.x;
    const int wave = tid >> 5;
    const int lane = tid & 31;
    const int half = lane >> 4;          // 0: lanes 0-15, 1: lanes 16-31
    const int l16  = lane & 15;

    if (tid < 32) {
        int gr = bm * 32 + tid;
        sn_s[tid] = (gr < N_DST) ? sampled_nodes[gr] : 0;
    }

    v8f acc[2][4] = {};

    // thread-invariant staging coordinates
    const int a_row0 = tid >> 4,          a_kk0 = tid & 15;     // A elems 0..255
    const int a_row1 = (tid + 256) >> 4,  a_kk1 = tid & 15;     // A elems 256..511
    const int a_gr0 = bm * 32 + a_row0, a_gr1 = bm * 32 + a_row1;

    for (int k0 = 0; k0 < NH; k0 += KC) {
        // ---- load phase: keep all global loads in flight ----
        float av0 = (a_gr0 < N_DST) ? agg[(size_t)a_gr0 * NH + k0 + a_kk0] : 0.f;
        float av1 = (a_gr1 < N_DST) ? agg[(size_t)a_gr1 * NH + k0 + a_kk1] : 0.f;
        float4 wv[8];
        #pragma unroll
        for (int j = 0; j < 8; ++j) {
            int i = tid + j * 256;
            int row = i >> 7, c4 = i & 127;
            wv[j] = ((const float4*)(W + (size_t)(k0 + row) * NH))[c4];
        }
        // ---- store phase ----
        As[a_row0 * AS_STRIDE + a_kk0] = av0;
        As[a_row1 * AS_STRIDE + a_kk1] = av1;
        #pragma unroll
        for (int j = 0; j < 8; ++j) {
            int i = tid + j * 256;
            int row = i >> 7, c4 = i & 127;
            float* d = &Bs[row * BS_STRIDE + c4 * 4];
            d[0] = wv[j].x; d[1] = wv[j].y; d[2] = wv[j].z; d[3] = wv[j].w;
        }
        __syncthreads();

        #pragma unroll
        for (int kk = 0; kk < KC; kk += 4) {
            int ak = kk + (half << 1);   // lanes 0-15: K=kk,kk+1 ; lanes 16-31: K=kk+2,kk+3
            v2f a0, a1;
            a0.x = As[l16 * AS_STRIDE + ak];
            a0.y = As[l16 * AS_STRIDE + ak + 1];
            a1.x = As[(l16 + 16) * AS_STRIDE + ak];
            a1.y = As[(l16 + 16) * AS_STRIDE + ak + 1];
            #pragma unroll
            for (int t = 0; t < 4; ++t) {
                int c = wave * 64 + t * 16 + l16;
                v2f b;
                b.x = Bs[ak * BS_STRIDE + c];
                b.y = Bs[(ak + 1) * BS_STRIDE + c];
                acc[0][t] = __builtin_amdgcn_wmma_f32_16x16x4_f32(
                    false, a0, false, b, (short)0, acc[0][t], false, false);
                acc[1][t] = __builtin_amdgcn_wmma_f32_16x16x4_f32(
                    false, a1, false, b, (short)0, acc[1][t], false, false);
            }
        }
        __syncthreads();
    }

    // epilogue: two passes of 16 rows through the Os slab (aliases Bs)
    float bc[4];
    #pragma unroll
    for (int t = 0; t < 4; ++t) bc[t] = bias[wave * 64 + t * 16 + l16];

    const float4* sc4 = (const float4*)scale;
    const float4* of4 = (const float4*)offset;

    #pragma unroll
    for (int p = 0; p < 2; ++p) {
        __syncthreads();   // p=0: redundant after K loop; p=1: protect Os reuse
        // pass p part 1: bias + momentum(y_buf gather) + ELU -> Os slab
        #pragma unroll
        for (int t = 0; t < 4; ++t) {
            int c = wave * 64 + t * 16 + l16;
            #pragma unroll
            for (int r = 0; r < 8; ++r) {
                int m = r + (half << 3);     // row within 16-row slab
                float v = acc[p][t][r] + bc[t];
                float y = y_buf[(size_t)sn_s[p * 16 + m] * NH + c];
                v = 0.9f * v + 0.1f * y;
                v = (v > 0.f) ? v : (expf(v) - 1.f);
                Os[m * OS_STRIDE + c] = v;
            }
        }
        __syncthreads();

        // pass p part 2: per-row layernorm; wave w handles slab rows 2w, 2w+1
        #pragma unroll
        for (int rr = 0; rr < 2; ++rr) {
            int m = wave * 2 + rr;
            float4 v[4];
            float s = 0.f, q = 0.f;
            #pragma unroll
            for (int c = 0; c < 4; ++c) {
                v[c] = *(const float4*)&Os[m * OS_STRIDE + c * 128 + lane * 4];
                s += v[c].x + v[c].y + v[c].z + v[c].w;
                q += v[c].x * v[c].x + v[c].y * v[c].y + v[c].z * v[c].z + v[c].w * v[c].w;
            }
            #pragma unroll
            for (int off = 16; off >= 1; off >>= 1) {
                s += __shfl_xor(s, off, 32);
                q += __shfl_xor(q, off, 32);
            }
            float mean = s * (1.f / 512.f);
            float var  = q * (1.f / 512.f) - mean * mean + 1e-9f;
            float rstd = rsqrtf(var);
            int gr = bm * 32 + p * 16 + m;
            if (gr < N_DST) {
                float4* orow = (float4*)(out + (size_t)gr * NH);
                #pragma unroll
                for (int c = 0; c < 4; ++c) {
                    float4 sc = sc4[c * 32 + lane];
                    float4 of = of4[c * 32 + lane];
                    float4 o;
                    o.x = (v[c].x - mean) * sc.x * rstd + of.x;
                    o.y = (v[c].y - mean) * sc.y * rstd + of.y;
                    o.z = (v[c].z - mean) * sc.z * rstd + of.z;
                    o.w = (v[c].w - mean) * sc.w * rstd + of.w;
                    orow[c * 32 + lane] = o;
                }
            }
        }
    }
}

// ---------------- launcher ----------------

extern "C" void kernel_launch(void* const* d_in, const int* in_sizes, int n_in,
                              void* d_out, int out_size, void* d_ws, size_t ws_size,
                              hipStream_t stream) {
    const float* x             = (const float*)d_in[0];
    const int*   adj_rows      = (const int*)d_in[1];
    const int*   adj_cols      = (const int*)d_in[2];
    const float* adj_vals      = (const float*)d_in[3];
    const int*   sampled_nodes = (const int*)d_in[4];
    const float* y_buf         = (const float*)d_in[5];
    const float* W             = (const float*)d_in[6];
    const float* bias          = (const float*)d_in[7];
    const float* scale         = (const float*)d_in[8];
    const float* offset        = (const float*)d_in[9];
    float* out = (float*)d_out;

    // workspace layout
    const size_t AGG_BYTES = (size_t)N_DST * NH * sizeof(float);   // 51,200,000 (256-aligned)
    float* agg     = (float*)d_ws;
    int*   row_ptr = (int*)((char*)d_ws + AGG_BYTES);              // N_DST+1 (padded to 25024)
    int*   fill    = row_ptr + 25024;                               // N_DST   (padded to 25024)
    int*   eidx    = fill + 25024;                                  // N_EDGES

    // 1. zero row_ptr + fill (50048 ints, contiguous)
    gc_zero_kernel<<<(50048 + 255) / 256, 256, 0, stream>>>(row_ptr, 50048);
    // 2. per-row edge counts into row_ptr[r+1]
    gc_hist_kernel<<<(N_EDGES + 255) / 256, 256, 0, stream>>>(adj_rows, row_ptr);
    // 3. inclusive scan -> CSR row pointers
    gc_scan_kernel<<<1, 1024, 0, stream>>>(row_ptr);
    // 4. scatter edge ids into CSR order
    gc_scatter_kernel<<<(N_EDGES + 255) / 256, 256, 0, stream>>>(adj_rows, row_ptr, fill, eidx);
    // 5. gather SpMM (no atomics on feature data)
    gc_spmm_kernel<<<N_DST, 128, 0, stream>>>(x, adj_cols, adj_vals, row_ptr, eidx, agg);
    // 6. fused WMMA-f32 GEMM (32x512 slab per block) + bias + momentum + ELU + layernorm
    gc_gemm_fused_kernel<<<(N_DST + 31) / 32, 256, 0, stream>>>(
        agg, W, bias, sampled_nodes, y_buf, scale, offset, out);
    (void)in_sizes; (void)n_in; (void)out_size; (void)ws_size;
}